// MinibatchDiscrimination_66391604461943
// MI455X (gfx1250) — compile-verified
//
#include <hip/hip_runtime.h>
#include <hip/hip_bf16.h>

// Problem dims (fixed by reference)
#define BATCH    256
#define IN_F     1024
#define OUT_F    128
#define KD       8
#define NCOL     (OUT_F * KD)   // 1024 = columns of M / W
#define OUTW     (IN_F + OUT_F) // 1152 = output row width

typedef float v2f __attribute__((ext_vector_type(2)));
typedef float v8f __attribute__((ext_vector_type(8)));

// ---------------------------------------------------------------------------
// Kernel 1: M = x @ W  where W = T.reshape(1024, 1024)   (f32 WMMA)
//
// One wave computes one 16x16 tile of M using V_WMMA_F32_16X16X4_F32.
// Fragment layouts per CDNA5 ISA 7.12.2:
//   A (16x4 f32):  lanes 0-15 -> M=lane, K={0,1} in v0,v1 ; lanes 16-31 -> K={2,3}
//   B (4x16 f32):  lanes 0-15 -> N=lane, K={0,1} in v0,v1 ; lanes 16-31 -> K={2,3}
//   C/D (16x16):   VGPR r: lanes 0-15 -> M=r, lanes 16-31 -> M=r+8, N=lane%16
// ---------------------------------------------------------------------------
__global__ __launch_bounds__(128)
void minibatch_gemm_wmma_f32(const float* __restrict__ A,   // BATCH x IN_F
                             const float* __restrict__ W,   // IN_F  x NCOL
                             float* __restrict__ M)         // BATCH x NCOL
{
    const int lane  = threadIdx.x & 31;
    const int wave  = threadIdx.x >> 5;                 // 4 waves / block
    const int tile  = blockIdx.x * 4 + wave;            // 1024 tiles total
    const int tilesPerRow = NCOL / 16;                  // 64
    const int row0  = (tile / tilesPerRow) * 16;        // M-tile row
    const int col0  = (tile % tilesPerRow) * 16;        // M-tile col

    const int half = lane >> 4;      // 0: K 0-1 / M 0-7,  1: K 2-3 / M 8-15
    const int lid  = lane & 15;

    v8f acc = {};

    // per-lane base pointers
    const float* arow = A + (size_t)(row0 + lid) * IN_F + half * 2; // A[m][kk+half*2+{0,1}]
    const float* bcol = W + (size_t)(half * 2) * NCOL + col0 + lid; // W[kk+half*2+{0,1}][n]

    for (int kk = 0; kk < IN_F; kk += 4) {
        v2f a;
        a.x = arow[kk + 0];
        a.y = arow[kk + 1];
        v2f b;
        b.x = bcol[(size_t)(kk + 0) * NCOL];
        b.y = bcol[(size_t)(kk + 1) * NCOL];
        // 8 args: (neg_a, A, neg_b, B, c_mod, C, reuse_a, reuse_b)
        acc = __builtin_amdgcn_wmma_f32_16x16x4_f32(
            false, a, false, b, (short)0, acc, false, false);
    }

    // Store D: VGPR r holds row (row0 + r + half*8), column col0 + lid
    float* dst = M + (size_t)(row0 + half * 8) * NCOL + col0 + lid;
#pragma unroll
    for (int r = 0; r < 8; ++r)
        dst[(size_t)r * NCOL] = acc[r];
}

// ---------------------------------------------------------------------------
// Kernel 2: c[i,f] = sum_j exp(-sum_k |M[j,f,k]-M[i,f,k]|) - 1
// One block per sample i (256 blocks), one thread per feature f (128 threads).
// At fixed j the 128 threads read a contiguous 4 KB row of M -> coalesced,
// L2-resident (M is 1 MB, L2 is 192 MB).
// ---------------------------------------------------------------------------
__global__ __launch_bounds__(128)
void minibatch_pairwise(const float* __restrict__ M,   // BATCH x NCOL
                        float* __restrict__ out)       // BATCH x OUTW
{
    const int i = blockIdx.x;       // sample
    const int f = threadIdx.x;      // feature

    const float4* mi4 = (const float4*)(M + (size_t)i * NCOL + f * KD);
    const float4 a0 = mi4[0];
    const float4 a1 = mi4[1];

    float acc = 0.0f;
#pragma unroll 4
    for (int j = 0; j < BATCH; ++j) {
        const float4* mj4 = (const float4*)(M + (size_t)j * NCOL + f * KD);
        const float4 b0 = mj4[0];
        const float4 b1 = mj4[1];
        float d = fabsf(a0.x - b0.x) + fabsf(a0.y - b0.y) +
                  fabsf(a0.z - b0.z) + fabsf(a0.w - b0.w) +
                  fabsf(a1.x - b1.x) + fabsf(a1.y - b1.y) +
                  fabsf(a1.z - b1.z) + fabsf(a1.w - b1.w);
        acc += __expf(-d);
    }
    // self term exp(0)=1 is included in acc; subtract it
    out[(size_t)i * OUTW + IN_F + f] = acc - 1.0f;
}

// ---------------------------------------------------------------------------
// Kernel 3: out[:, :1024] = x  (vectorized copy)
// ---------------------------------------------------------------------------
__global__ __launch_bounds__(256)
void minibatch_copy_x(const float* __restrict__ x,     // BATCH x IN_F
                      float* __restrict__ out)         // BATCH x OUTW
{
    const int i = blockIdx.x;
    const int t = threadIdx.x;                         // 256 threads * float4 = 1024 floats
    const float4* src = (const float4*)(x + (size_t)i * IN_F);
    float4* dst = (float4*)(out + (size_t)i * OUTW);
    dst[t] = src[t];
}

extern "C" void kernel_launch(void* const* d_in, const int* in_sizes, int n_in,
                              void* d_out, int out_size, void* d_ws, size_t ws_size,
                              hipStream_t stream) {
    const float* x = (const float*)d_in[0];           // BATCH x IN_F
    const float* T = (const float*)d_in[1];           // IN_F x OUT_F x KD == IN_F x NCOL
    float* out = (float*)d_out;                       // BATCH x OUTW
    float* M   = (float*)d_ws;                        // BATCH x NCOL scratch (1 MB)

    (void)in_sizes; (void)n_in; (void)out_size; (void)ws_size;

    // 1024 tiles, 4 waves (of 32) per block -> 256 blocks of 128 threads
    minibatch_gemm_wmma_f32<<<256, 128, 0, stream>>>(x, T, M);
    // one block per sample, one thread per feature
    minibatch_pairwise<<<BATCH, OUT_F, 0, stream>>>(M, out);
    // copy x into left part of output
    minibatch_copy_x<<<BATCH, 256, 0, stream>>>(x, out);
}